// VanillaAttention_19619410608256
// MI455X (gfx1250) — compile-verified
//
#include <hip/hip_runtime.h>
#include <math.h>

typedef __attribute__((ext_vector_type(2))) float v2f;
typedef __attribute__((ext_vector_type(4))) float v4f;
typedef __attribute__((ext_vector_type(8))) float v8f;

#define BATCH 16
#define LSEQ  8192
#define DIM   512
#define NEG_INF_F (-1e30f)
#define CHUNKS_PER_B 64
#define ROWS_PER_CHUNK (LSEQ / CHUNKS_PER_B) /* 128 */

// ---------------------------------------------------------------------------
// Kernel 1: Y[16,512] = X[16,512] @ W^T + bias   (torch Linear, W is [out,in])
// One wave per 16x16 output tile, V_WMMA_F32_16X16X4_F32, K-loop of 128.
// Fragment layouts per CDNA5 ISA 7.12.2 (32-bit A 16x4, C/D 16x16).
// ---------------------------------------------------------------------------
__global__ __launch_bounds__(128) void proj_kernel(
    const float* __restrict__ X, const float* __restrict__ W,
    const float* __restrict__ bias, float* __restrict__ Y) {
  const int lane = threadIdx.x & 31;
  const int wid  = blockIdx.x * (blockDim.x >> 5) + (threadIdx.x >> 5);
  const int n0   = wid * 16;                  // output-column tile base
  const int mr   = lane & 15;                 // A row / B column within tile
  const int kb   = (lane >> 4) << 1;          // K sub-offset: 0 lanes 0-15, 2 lanes 16-31

  v8f c = {};
  for (int k = 0; k < DIM; k += 4) {
    v2f a, b;
    a.x = X[mr * DIM + k + kb];               // A[m][k+kb]
    a.y = X[mr * DIM + k + kb + 1];           // A[m][k+kb+1]
    // B[kk][n] = W^T[kk][n] = W[n][kk]
    b.x = W[(n0 + mr) * DIM + k + kb];
    b.y = W[(n0 + mr) * DIM + k + kb + 1];
    c = __builtin_amdgcn_wmma_f32_16x16x4_f32(false, a, false, b, (short)0, c,
                                              false, false);
  }
  const int col     = n0 + (lane & 15);
  const int rowbase = (lane >> 4) << 3;       // lanes 16-31 hold rows 8..15
  const float bb    = bias[col];
#pragma unroll
  for (int i = 0; i < 8; ++i)
    Y[(rowbase + i) * DIM + col] = c[i] + bb;
}

// ---------------------------------------------------------------------------
// Kernel 4: out = tanh(ctx @ Wv^T + bv + q @ Wq^T + bq) — two WMMA GEMMs
// accumulated into the same C tile.
// ---------------------------------------------------------------------------
__global__ __launch_bounds__(128) void out_kernel(
    const float* __restrict__ ctx, const float* __restrict__ Wv,
    const float* __restrict__ bv, const float* __restrict__ q,
    const float* __restrict__ Wq, const float* __restrict__ bq,
    float* __restrict__ out) {
  const int lane = threadIdx.x & 31;
  const int wid  = blockIdx.x * (blockDim.x >> 5) + (threadIdx.x >> 5);
  const int n0   = wid * 16;
  const int mr   = lane & 15;
  const int kb   = (lane >> 4) << 1;

  v8f c = {};
  for (int k = 0; k < DIM; k += 4) {
    v2f a1, b1, a2, b2;
    a1.x = ctx[mr * DIM + k + kb];
    a1.y = ctx[mr * DIM + k + kb + 1];
    b1.x = Wv[(n0 + mr) * DIM + k + kb];
    b1.y = Wv[(n0 + mr) * DIM + k + kb + 1];
    c = __builtin_amdgcn_wmma_f32_16x16x4_f32(false, a1, false, b1, (short)0, c,
                                              false, false);
    a2.x = q[mr * DIM + k + kb];
    a2.y = q[mr * DIM + k + kb + 1];
    b2.x = Wq[(n0 + mr) * DIM + k + kb];
    b2.y = Wq[(n0 + mr) * DIM + k + kb + 1];
    c = __builtin_amdgcn_wmma_f32_16x16x4_f32(false, a2, false, b2, (short)0, c,
                                              false, false);
  }
  const int col     = n0 + (lane & 15);
  const int rowbase = (lane >> 4) << 3;
  const float bsum  = bv[col] + bq[col];
#pragma unroll
  for (int i = 0; i < 8; ++i)
    out[(rowbase + i) * DIM + col] = tanhf(c[i] + bsum);
}

// ---------------------------------------------------------------------------
// Kernel 2: single streaming pass over `value` (flash-attention style).
// Each block: one (batch, 128-row chunk). 8 waves, each wave owns 16 rows.
// Per wave: running max m, running sum s, running ctx[512] (16 VGPRs/lane).
// Non-temporal 128-bit loads: value is stream-once, keep it out of caches.
// ---------------------------------------------------------------------------
__device__ inline float v4dot(v4f a, v4f b) {
  return a.x * b.x + a.y * b.y + a.z * b.z + a.w * b.w;
}

__global__ __launch_bounds__(256) void flash_kernel(
    const float* __restrict__ value, const int* __restrict__ mask,
    const float* __restrict__ q, float* __restrict__ pm,
    float* __restrict__ ps, float* __restrict__ pctx) {
  __shared__ v4f lq4[DIM / 4];
  __shared__ float red_m[8];
  __shared__ float red_s[8];
  __shared__ v4f red_ctx[8][DIM / 4];

  const int b     = blockIdx.x / CHUNKS_PER_B;
  const int chunk = blockIdx.x % CHUNKS_PER_B;
  const int t     = threadIdx.x;
  const int wave  = t >> 5;
  const int lane  = t & 31;

  // Stage q[b,:] into LDS (2 floats per thread).
  ((float*)lq4)[t]       = q[b * DIM + t];
  ((float*)lq4)[t + 256] = q[b * DIM + t + 256];
  __syncthreads();

  const v4f q0 = lq4[lane], q1 = lq4[lane + 32], q2 = lq4[lane + 64],
            q3 = lq4[lane + 96];

  v4f a0 = {0.f, 0.f, 0.f, 0.f}, a1 = a0, a2 = a0, a3 = a0;
  float m = -3.0e38f, ssum = 0.0f;
  const int lbase = chunk * ROWS_PER_CHUNK;

  for (int rr = 0; rr < ROWS_PER_CHUNK / 8; ++rr) {
    const int l = lbase + wave + rr * 8;
    const v4f* vr = (const v4f*)(value + ((size_t)b * LSEQ + l) * DIM);
    const v4f v0 = __builtin_nontemporal_load(&vr[lane]);
    const v4f v1 = __builtin_nontemporal_load(&vr[lane + 32]);
    const v4f v2 = __builtin_nontemporal_load(&vr[lane + 64]);
    const v4f v3 = __builtin_nontemporal_load(&vr[lane + 96]);

    float local = v4dot(v0, q0) + v4dot(v1, q1) + v4dot(v2, q2) + v4dot(v3, q3);
#pragma unroll
    for (int o = 16; o > 0; o >>= 1) local += __shfl_xor(local, o, 32);

    const float s  = (mask[b * LSEQ + l] != 0) ? NEG_INF_F : local;
    const float nm = fmaxf(m, s);
    const float sc = __expf(m - nm);   // == 0 on first iteration (m = -3e38)
    const float p  = __expf(s - nm);
    a0 = a0 * sc + v0 * p;
    a1 = a1 * sc + v1 * p;
    a2 = a2 * sc + v2 * p;
    a3 = a3 * sc + v3 * p;
    ssum = ssum * sc + p;
    m = nm;
  }

  // Per-wave partials -> LDS
  red_ctx[wave][lane]      = a0;
  red_ctx[wave][lane + 32] = a1;
  red_ctx[wave][lane + 64] = a2;
  red_ctx[wave][lane + 96] = a3;
  if (lane == 0) { red_m[wave] = m; red_s[wave] = ssum; }
  __syncthreads();

  // Wave 0 combines the 8 wave-partials into one block partial.
  if (wave == 0) {
    float M = red_m[0];
#pragma unroll
    for (int w = 1; w < 8; ++w) M = fmaxf(M, red_m[w]);
    float S = 0.0f;
    v4f o0 = {0.f, 0.f, 0.f, 0.f}, o1 = o0, o2 = o0, o3 = o0;
#pragma unroll
    for (int w = 0; w < 8; ++w) {
      const float e = __expf(red_m[w] - M);
      S += e * red_s[w];
      o0 += red_ctx[w][lane] * e;
      o1 += red_ctx[w][lane + 32] * e;
      o2 += red_ctx[w][lane + 64] * e;
      o3 += red_ctx[w][lane + 96] * e;
    }
    if (lane == 0) { pm[blockIdx.x] = M; ps[blockIdx.x] = S; }
    v4f* pc = (v4f*)(pctx + (size_t)blockIdx.x * DIM);
    pc[lane]      = o0;
    pc[lane + 32] = o1;
    pc[lane + 64] = o2;
    pc[lane + 96] = o3;
  }
}

// ---------------------------------------------------------------------------
// Kernel 3: per batch, merge 64 chunk partials and normalize -> context[b,:]
// ---------------------------------------------------------------------------
__global__ __launch_bounds__(256) void combine_kernel(
    const float* __restrict__ pm, const float* __restrict__ ps,
    const float* __restrict__ pctx, float* __restrict__ ctx) {
  const int b    = blockIdx.x;
  const int t    = threadIdx.x;
  const int base = b * CHUNKS_PER_B;

  float M = -3.0e38f;
  for (int c = 0; c < CHUNKS_PER_B; ++c) M = fmaxf(M, pm[base + c]);
  float den = 0.0f;
  for (int c = 0; c < CHUNKS_PER_B; ++c)
    den += __expf(pm[base + c] - M) * ps[base + c];
  const float inv = 1.0f / den;

  const int d0 = t * 2;
  float s0 = 0.f, s1 = 0.f;
  for (int c = 0; c < CHUNKS_PER_B; ++c) {
    const float e = __expf(pm[base + c] - M);
    const float* pc = pctx + (size_t)(base + c) * DIM;
    s0 += e * pc[d0];
    s1 += e * pc[d0 + 1];
  }
  ctx[b * DIM + d0]     = s0 * inv;
  ctx[b * DIM + d0 + 1] = s1 * inv;
}

// ---------------------------------------------------------------------------
extern "C" void kernel_launch(void* const* d_in, const int* in_sizes, int n_in,
                              void* d_out, int out_size, void* d_ws,
                              size_t ws_size, hipStream_t stream) {
  const float* query   = (const float*)d_in[0];
  const float* value   = (const float*)d_in[1];
  const int*   mask    = (const int*)d_in[2];
  const float* W_align = (const float*)d_in[3];
  const float* b_align = (const float*)d_in[4];
  const float* W_query = (const float*)d_in[5];
  const float* b_query = (const float*)d_in[6];
  const float* W_value = (const float*)d_in[7];
  const float* b_value = (const float*)d_in[8];
  float* out = (float*)d_out;

  float* ws   = (float*)d_ws;
  float* q    = ws;               // 16*512          = 8192 floats
  float* ctx  = ws + 8192;        // 16*512          = 8192 floats
  float* pm   = ws + 16384;       // 1024 floats
  float* ps   = ws + 17408;       // 1024 floats
  float* pctx = ws + 18432;       // 1024*512 floats (16B-aligned offset)

  // 1) q = query @ W_align^T + b_align   (WMMA f32)
  proj_kernel<<<8, 128, 0, stream>>>(query, W_align, b_align, q);
  // 2) single streaming pass over value: masked score + online softmax + ctx
  flash_kernel<<<BATCH * CHUNKS_PER_B, 256, 0, stream>>>(value, mask, q, pm,
                                                         ps, pctx);
  // 3) merge partials per batch -> context
  combine_kernel<<<BATCH, 256, 0, stream>>>(pm, ps, pctx, ctx);
  // 4) out = tanh(ctx @ Wv^T + bv + q @ Wq^T + bq)   (WMMA f32 x2)
  out_kernel<<<8, 128, 0, stream>>>(ctx, W_value, b_value, q, W_query, b_query,
                                    out);
}